// GCN_78580721647789
// MI455X (gfx1250) — compile-verified
//
#include <hip/hip_runtime.h>
#include <hip/hip_bf16.h>
#include <math.h>

typedef float v2f __attribute__((ext_vector_type(2)));
typedef float v8f __attribute__((ext_vector_type(8)));

// ---------------------------------------------------------------------------
// Degree / normalization kernels
// ---------------------------------------------------------------------------
__global__ void k_fill(float* __restrict__ p, float val, int n) {
  int i = blockIdx.x * blockDim.x + threadIdx.x;
  if (i < n) p[i] = val;
}

__global__ void k_deg_edges(const int* __restrict__ dst, float* __restrict__ deg, int E) {
  int e = blockIdx.x * blockDim.x + threadIdx.x;
  if (e < E) unsafeAtomicAdd(&deg[dst[e]], 1.0f);
}

__global__ void k_rsqrt(const float* __restrict__ deg, float* __restrict__ dinv, int n) {
  int i = blockIdx.x * blockDim.x + threadIdx.x;
  if (i < n) dinv[i] = rsqrtf(deg[i]);
}

// ---------------------------------------------------------------------------
// Pack B (K=128 x N) into WMMA-fragment order, zero-padded past N:
//   BF[((nt*32 + t)*32 + lane)] = { B[4t+2*half][nt*16+lm], B[4t+2*half+1][...] }
// where half = lane>>4, lm = lane&15. One float2 per (tile, kstep, lane).
// ---------------------------------------------------------------------------
__global__ void k_pack_bfrag(const float* __restrict__ B, v2f* __restrict__ BF,
                             int N, int nTiles) {
  int i = blockIdx.x * blockDim.x + threadIdx.x;
  int total = nTiles * 1024;                     // 32 ksteps * 32 lanes per tile
  if (i >= total) return;
  int lane = i & 31;
  int t    = (i >> 5) & 31;
  int nt   = i >> 10;
  int half = lane >> 4;
  int lm   = lane & 15;
  int col  = nt * 16 + lm;
  int kk   = 4 * t + 2 * half;
  v2f v;
  v.x = (col < N) ? B[(size_t)kk * N + col]       : 0.0f;
  v.y = (col < N) ? B[(size_t)(kk + 1) * N + col] : 0.0f;
  BF[i] = v;
}

// ---------------------------------------------------------------------------
// FP32 WMMA GEMM:  C[M,N] = A[M,K=128] * B[K=128,N]  (B pre-packed as BF)
// grid.x = M/16, block = 256 = 8 waves. Wave handles N-tiles wave, wave+8, ...
// Inner loop per kstep: ds_load_b64 (A frag) + global_load_b64 (B frag) + WMMA.
// ---------------------------------------------------------------------------
__global__ __launch_bounds__(256) void gcn_gemm_wmma_f32(
    const float* __restrict__ A, const v2f* __restrict__ BF,
    float* __restrict__ C, int M, int N, int nTiles) {
  constexpr int K = 128;
  constexpr int LDS_STRIDE = 132;                // 132 mod 64 = 4 -> conflict-free
  __shared__ float As[16 * LDS_STRIDE];

  const int m0  = blockIdx.x * 16;
  const int tid = threadIdx.x;

  // Cooperative load of the 16x128 A tile (2048 floats / 256 threads = 8 each)
#pragma unroll
  for (int j = 0; j < 8; ++j) {
    int off = tid + j * 256;
    int row = off >> 7;                          // /128
    int col = off & 127;
    As[row * LDS_STRIDE + col] = A[(size_t)(m0 + row) * K + col];
  }
  __syncthreads();

  const int wave = tid >> 5;
  const int lane = tid & 31;
  const int half = lane >> 4;                    // 0: lanes 0-15, 1: lanes 16-31
  const int lm   = lane & 15;

  // A fragment base: lm*132 + 2*half is even -> 8-byte aligned float2 reads
  const float* aBase = &As[lm * LDS_STRIDE + 2 * half];

  for (int nt = wave; nt < nTiles; nt += 8) {
    const v2f* bBase = BF + (size_t)nt * 1024 + lane;
    v8f c = {};
#pragma unroll
    for (int t = 0; t < 32; ++t) {
      v2f a = *(const v2f*)(aBase + 4 * t);      // ds_load_b64, immediate offset
      v2f b = bBase[(size_t)t * 32];             // global_load_b64, immediate offset
      c = __builtin_amdgcn_wmma_f32_16x16x4_f32(
          /*neg_a=*/false, a, /*neg_b=*/false, b,
          /*c_mod=*/(short)0, c, /*reuse_a=*/false, /*reuse_b=*/false);
    }
    const int col = nt * 16 + lm;
    if (col < N) {
      const int mBase = m0 + 8 * half;           // lanes 0-15 -> rows r, 16-31 -> r+8
#pragma unroll
      for (int r = 0; r < 8; ++r)
        C[(size_t)(mBase + r) * N + col] = c[r];
    }
  }
}

// ---------------------------------------------------------------------------
// Aggregation:  out[i] = dinv[i]^2 * h[i]  (self-loop)  then edge atomics
// ---------------------------------------------------------------------------
__global__ void k_self_init(const float* __restrict__ h, const float* __restrict__ dinv,
                            float* __restrict__ out, int total, int F) {
  int i = blockIdx.x * blockDim.x + threadIdx.x;
  if (i >= total) return;
  int node = i / F;
  float di = dinv[node];
  out[i] = di * di * h[i];
}

// 128 threads (4 waves) handle one edge's 128 features (tid>>7 uniform across a wave32).
__global__ void k_edge_f128(const float* __restrict__ h, const int* __restrict__ src,
                            const int* __restrict__ dst, const float* __restrict__ dinv,
                            float* __restrict__ out, int E) {
  int tid = blockIdx.x * blockDim.x + threadIdx.x;
  int e = tid >> 7;
  if (e >= E) return;
  int f = tid & 127;
  int s = src[e], d = dst[e];
  float coeff = dinv[s] * dinv[d];
  unsafeAtomicAdd(&out[(size_t)d * 128 + f], coeff * h[(size_t)s * 128 + f]);
}

// 64 lanes per edge, 40 active features.
__global__ void k_edge_f40(const float* __restrict__ h, const int* __restrict__ src,
                           const int* __restrict__ dst, const float* __restrict__ dinv,
                           float* __restrict__ out, int E) {
  int tid = blockIdx.x * blockDim.x + threadIdx.x;
  int e = tid >> 6;
  int f = tid & 63;
  if (e >= E || f >= 40) return;
  int s = src[e], d = dst[e];
  float coeff = dinv[s] * dinv[d];
  unsafeAtomicAdd(&out[(size_t)d * 40 + f], coeff * h[(size_t)s * 40 + f]);
}

__global__ void k_bias_relu(float* __restrict__ out, const float* __restrict__ b,
                            int total, int F) {
  int i = blockIdx.x * blockDim.x + threadIdx.x;
  if (i >= total) return;
  float v = out[i] + b[i % F];
  out[i] = v > 0.0f ? v : 0.0f;
}

// ---------------------------------------------------------------------------
// Fused bias + log_softmax over 40 classes; one wave32 per row.
// ---------------------------------------------------------------------------
__global__ __launch_bounds__(256) void k_logsoftmax40(float* __restrict__ out,
                                                      const float* __restrict__ b, int N) {
  int row  = blockIdx.x * 8 + (threadIdx.x >> 5);
  int lane = threadIdx.x & 31;
  if (row >= N) return;                          // wave-uniform exit
  float* p = out + (size_t)row * 40;
  float v0 = p[lane] + b[lane];                  // lane < 32 < 40 always valid
  bool has2 = (lane + 32) < 40;                  // lanes 0..7
  float v1 = has2 ? p[lane + 32] + b[lane + 32] : -INFINITY;

  float m = fmaxf(v0, v1);
#pragma unroll
  for (int off = 16; off > 0; off >>= 1) m = fmaxf(m, __shfl_xor(m, off, 32));

  float s = expf(v0 - m) + (has2 ? expf(v1 - m) : 0.0f);
#pragma unroll
  for (int off = 16; off > 0; off >>= 1) s += __shfl_xor(s, off, 32);

  float lse = logf(s);
  p[lane] = v0 - m - lse;
  if (has2) p[lane + 32] = v1 - m - lse;
}

// ---------------------------------------------------------------------------
// Host orchestration
// ---------------------------------------------------------------------------
static inline int cdiv_ll(long long a, long long b) { return (int)((a + b - 1) / b); }

extern "C" void kernel_launch(void* const* d_in, const int* in_sizes, int n_in,
                              void* d_out, int out_size, void* d_ws, size_t ws_size,
                              hipStream_t stream) {
  const float* x  = (const float*)d_in[0];
  const int*   ei = (const int*)  d_in[1];
  const float* W1 = (const float*)d_in[2];
  const float* b1 = (const float*)d_in[3];
  const float* W2 = (const float*)d_in[4];
  const float* b2 = (const float*)d_in[5];

  const int F = 128, H = 128, C = 40;
  const int N = in_sizes[0] / F;                 // 50000 (multiple of 16)
  const int E = in_sizes[1] / 2;                 // 800000
  const int* src = ei;                           // edge_index[0]
  const int* dst = ei + E;                       // edge_index[1]

  const int nT1 = (H + 15) / 16;                 // 8
  const int nT2 = (C + 15) / 16;                 // 3

  float* ws   = (float*)d_ws;
  float* deg  = ws;                 ws += N;
  float* dinv = ws;                 ws += N;
  float* h1   = ws;                 ws += (size_t)N * H;
  float* o1   = ws;                 ws += (size_t)N * H;
  float* h2   = ws;                 ws += (size_t)N * C;
  v2f* bf1    = (v2f*)ws;           ws += (size_t)nT1 * 1024 * 2;
  v2f* bf2    = (v2f*)ws;           ws += (size_t)nT2 * 1024 * 2;
  float* out  = (float*)d_out;

  const int B = 256;

  // Symmetric normalization: deg includes self-loop (init 1.0)
  k_fill      <<<cdiv_ll(N, B), B, 0, stream>>>(deg, 1.0f, N);
  k_deg_edges <<<cdiv_ll(E, B), B, 0, stream>>>(dst, deg, E);
  k_rsqrt     <<<cdiv_ll(N, B), B, 0, stream>>>(deg, dinv, N);

  // Pack weight matrices into WMMA fragment order (tiny, done per call)
  k_pack_bfrag<<<cdiv_ll((long long)nT1 * 1024, B), B, 0, stream>>>(W1, bf1, H, nT1);
  k_pack_bfrag<<<cdiv_ll((long long)nT2 * 1024, B), B, 0, stream>>>(W2, bf2, C, nT2);

  // Layer 1: h1 = x @ W1 ; aggregate ; + b1 ; relu
  gcn_gemm_wmma_f32<<<N / 16, B, 0, stream>>>(x, bf1, h1, N, H, nT1);
  k_self_init <<<cdiv_ll((long long)N * H, B), B, 0, stream>>>(h1, dinv, o1, N * H, H);
  k_edge_f128 <<<cdiv_ll((long long)E * 128, B), B, 0, stream>>>(h1, src, dst, dinv, o1, E);
  k_bias_relu <<<cdiv_ll((long long)N * H, B), B, 0, stream>>>(o1, b1, N * H, H);

  // Layer 2: h2 = o1 @ W2 ; aggregate into d_out ; + b2 ; log_softmax
  gcn_gemm_wmma_f32<<<N / 16, B, 0, stream>>>(o1, bf2, h2, N, C, nT2);
  k_self_init <<<cdiv_ll((long long)N * C, B), B, 0, stream>>>(h2, dinv, out, N * C, C);
  k_edge_f40  <<<cdiv_ll((long long)E * 64, B), B, 0, stream>>>(h2, src, dst, dinv, out, E);
  k_logsoftmax40<<<cdiv_ll(N, 8), B, 0, stream>>>(out, b2, N);
}